// FixedRadiusSearch_49581102465309
// MI455X (gfx1250) — compile-verified
//
#include <hip/hip_runtime.h>

typedef __attribute__((ext_vector_type(2))) float v2f;
typedef __attribute__((ext_vector_type(8))) float v8f;

#define NPTS       32768
#define NQ         32768
#define KMAX       64
#define CHUNK_PTS  2048       // 2048 * 16B = 32KB LDS stage
#define THREADS    256        // 8 waves; each wave owns 16 queries -> 128 q/block

// ---------------------------------------------------------------------------
// Kernel 1: fixed-radius search via V_WMMA_F32_16X16X4_F32 distance filter.
//   A (16x4, per wave)  = -2 * [qx qy qz 0]   (K padded 3->4)
//   B (4x16, per tile)  =      [px py pz 0]^T
//   C preloaded with |q|^2_row + |p|^2_col  ==> WMMA output = squared L2 dist
// ---------------------------------------------------------------------------
__global__ __launch_bounds__(THREADS) void frs_wmma_kernel(
    const float* __restrict__ points, const float* __restrict__ queries,
    const float* __restrict__ radius_p, int* __restrict__ out_idx,
    float* __restrict__ out_dist, int* __restrict__ counts)
{
    __shared__ float4 sPts[CHUNK_PTS];

    const int tid  = threadIdx.x;
    const int lane = tid & 31;
    const int wave = tid >> 5;
    const int qb   = (blockIdx.x * 8 + wave) * 16;   // this wave's first query
    const float r  = radius_p[0];
    const float r2 = r * r;

    // Per-lane query registers: lane L holds query row (L & 15).
    const int   qm  = qb + (lane & 15);
    const float qx  = queries[qm * 3 + 0];
    const float qy  = queries[qm * 3 + 1];
    const float qz  = queries[qm * 3 + 2];
    const float qsq = qx * qx + qy * qy + qz * qz;
    const bool  hi  = lane >= 16;

    // WMMA A operand (16x4 f32): lanes 0-15 -> K=0,1 ; lanes 16-31 -> K=2,3
    v2f a;
    a.x = hi ? (-2.0f * qz) : (-2.0f * qx);
    a.y = hi ? 0.0f        : (-2.0f * qy);

    // qrow[r] = |q|^2 of C-row r (low-half lanes) / r+8 (high-half lanes)
    float qrow[8];
#pragma unroll
    for (int rr = 0; rr < 8; ++rr) {
        int src = hi ? (rr + 8) : rr;
        int v = __builtin_amdgcn_ds_bpermute(src << 2, __float_as_int(qsq));
        qrow[rr] = __int_as_float(v);
    }

    int cnt = 0;
    const long long obase = (long long)qm * KMAX;    // owned by lanes 0..15

    for (int cb = 0; cb < NPTS; cb += CHUNK_PTS) {
        __syncthreads();
        // Cooperative stage: {x,y,z,|p|^2} per point into LDS.
#pragma unroll
        for (int i = 0; i < CHUNK_PTS / THREADS; ++i) {
            int pl = tid + i * THREADS;
            int pg = cb + pl;
            float px = points[pg * 3 + 0];
            float py = points[pg * 3 + 1];
            float pz = points[pg * 3 + 2];
            sPts[pl] = make_float4(px, py, pz, px * px + py * py + pz * pz);
            if (cb + CHUNK_PTS < NPTS)   // pull next chunk toward the WGP
                __builtin_prefetch(&points[(pg + CHUNK_PTS) * 3], 0, 1);
        }
        __syncthreads();

        for (int tb = 0; tb < CHUNK_PTS; tb += 16) {
            float4 p = sPts[tb + (lane & 15)];

            // B operand (4x16 f32): VGPR0 rows K=0 (lo lanes) / K=2 (hi lanes),
            //                       VGPR1 rows K=1 (lo lanes) / K=3 (hi lanes)
            v2f b;
            b.x = hi ? p.z : p.x;
            b.y = hi ? 0.0f : p.y;

            v8f c;
#pragma unroll
            for (int rr = 0; rr < 8; ++rr) c[rr] = qrow[rr] + p.w;

            // c = A*B + c  ==> squared L2 distances for 16q x 16p tile
            c = __builtin_amdgcn_wmma_f32_16x16x4_f32(
                    false, a, false, b, (short)0, c, false, false);

            unsigned bl[8];
            unsigned any = 0;
#pragma unroll
            for (int rr = 0; rr < 8; ++rr) {
                bl[rr] = __builtin_amdgcn_ballot_w32(c[rr] <= r2);
                any |= bl[rr];
            }

            if (any) {                      // wave-uniform rare path
                if (lane < 16) {
                    int rsel = lane & 7;
                    unsigned s01 = (rsel & 1) ? bl[1] : bl[0];
                    unsigned s23 = (rsel & 1) ? bl[3] : bl[2];
                    unsigned s45 = (rsel & 1) ? bl[5] : bl[4];
                    unsigned s67 = (rsel & 1) ? bl[7] : bl[6];
                    unsigned s03 = (rsel & 2) ? s23 : s01;
                    unsigned s47 = (rsel & 2) ? s67 : s45;
                    unsigned sel = (rsel & 4) ? s47 : s03;
                    unsigned mask = (lane < 8) ? (sel & 0xffffu) : (sel >> 16);
                    while (mask) {
                        int t = __builtin_ctz(mask);
                        mask &= mask - 1;
                        float4 pp = sPts[tb + t];
                        float d = qsq + pp.w -
                                  2.0f * (qx * pp.x + qy * pp.y + qz * pp.z);
                        d = fmaxf(d, 0.0f);
                        if (cnt < KMAX) {
                            out_idx[obase + cnt]  = cb + tb + t;
                            out_dist[obase + cnt] = d;
                            ++cnt;
                        }
                    }
                }
            }
        }
    }

    if (lane < 16) {
        counts[qm] = cnt;                     // already clipped at KMAX
        for (int j = cnt; j < KMAX; ++j) {    // pad (harness poisons d_out)
            out_idx[obase + j]  = -1;
            out_dist[obase + j] = 0.0f;
        }
    }
}

// ---------------------------------------------------------------------------
// Kernel 2: exclusive prefix sum of counts -> row_splits[Q+1]
// ---------------------------------------------------------------------------
__global__ __launch_bounds__(256) void frs_scan_kernel(
    const int* __restrict__ counts, int* __restrict__ row_splits)
{
    __shared__ int part[256];
    const int t = threadIdx.x;
    const int per = NQ / 256;                 // 128
    const int base = t * per;

    int s = 0;
    for (int i = 0; i < per; ++i) s += counts[base + i];
    part[t] = s;
    __syncthreads();

    for (int off = 1; off < 256; off <<= 1) {
        int v = 0;
        if (t >= off) v = part[t - off];
        __syncthreads();
        part[t] += v;
        __syncthreads();
    }

    int run = part[t] - s;                    // exclusive offset of this block
    for (int i = 0; i < per; ++i) {
        row_splits[base + i] = run;
        run += counts[base + i];
    }
    if (t == 255) row_splits[NQ] = run;       // grand total
}

// ---------------------------------------------------------------------------
extern "C" void kernel_launch(void* const* d_in, const int* in_sizes, int n_in,
                              void* d_out, int out_size, void* d_ws, size_t ws_size,
                              hipStream_t stream)
{
    (void)n_in; (void)out_size; (void)ws_size;
    const float* points  = (const float*)d_in[0];
    const float* queries = (const float*)d_in[1];
    const float* radius  = (const float*)d_in[2];
    const int Q = in_sizes[1] / 3;            // 32768

    // d_out layout (return order, all 4-byte elements):
    //   [Q*K] int32 neighbors_index | [Q+1] int32 row_splits | [Q*K] f32 dists
    int*   out_idx    = (int*)d_out;
    int*   row_splits = out_idx + (size_t)Q * KMAX;
    float* out_dist   = (float*)(row_splits + Q + 1);
    int*   counts     = (int*)d_ws;           // Q ints of scratch

    frs_wmma_kernel<<<Q / 128, THREADS, 0, stream>>>(
        points, queries, radius, out_idx, out_dist, counts);
    frs_scan_kernel<<<1, 256, 0, stream>>>(counts, row_splits);
}